// HBVModel_83872121356579
// MI455X (gfx1250) — compile-verified
//
#include <hip/hip_runtime.h>
#include <stdint.h>

// HBV model: 8192 independent scalar recurrences of length T=2048.
// One lane per (b,u); 256 single-wave workgroups; async global->LDS
// double-buffered input prefetch (CDNA5 ASYNCcnt path); latency-minimized
// serial update chain.

#define B_    128
#define T_    2048
#define U_    64
#define CHUNK 32
#define NCHUNK (T_ / CHUNK)
#define CLIPV 100000.0f

__global__ __launch_bounds__(32) void hbv_scan_kernel(
    const float* __restrict__ inputs,
    const float* __restrict__ kv, const float* __restrict__ fc,
    const float* __restrict__ lp, const float* __restrict__ cf,
    const float* __restrict__ bt, const float* __restrict__ ka,
    const float* __restrict__ ab, const float* __restrict__ lu,
    const float* __restrict__ ag, const float* __restrict__ kp,
    float* __restrict__ out)
{
    // [buf][t][0..31] = vi for this wave's 32 u-columns, [32..63] = ep. 16 KB.
    __shared__ float lds[2][CHUNK][U_];

    const int lane = (int)threadIdx.x;        // 0..31 (wave32)
    const int b    = (int)blockIdx.x >> 1;    // batch row
    const int u0   = ((int)blockIdx.x & 1) * 32;
    const int u    = u0 + lane;

    // ---- per-unit parameters, rescaled per reference _rescale ----
    const float kv_ = 1.75f   * kv[u] + 0.25f;
    const float fc_ = 950.0f  * fc[u] + 50.0f;
    const float lp_ = 0.999f  * lp[u] + 0.001f;
    const float cf_ = 2.99f   * cf[u] + 0.01f;
    const float bt_ = bt[u];
    const float ka_ = 0.4f    * ka[u] + 0.1f;
    const float ab_ = 0.1f    * ab[u] + 0.5f;
    const float lu_ = 49.999f * lu[u] + 0.001f;
    const float ag_ = 0.31f   * ag[u] + 0.02f;
    const float kp_ = 3.999f  * kp[u] + 0.001f;

    // hoisted invariants (shorten the serial dependence chain)
    const float tm     = lp_ * fc_;
    const float inv_tm = 1.0f / tm;
    const float inv_fc = 1.0f / fc_;
    const float cfofc  = cf_ * inv_fc;        // ca = max(cf - cfofc*sm, 0)
    const float kaab   = ka_ * ab_;
    const float kaabag = kaab * ag_;

    // ---- async copy address setup ----
    // Each b128 issue: 32 lanes x 16B = 512B = 2 timestep half-rows.
    // lane 0-7: vi row t | 8-15: ep row t | 16-23: vi row t+1 | 24-31: ep row t+1
    const unsigned long long gbase =
        (unsigned long long)(uintptr_t)(inputs + (size_t)b * T_ * 2 * U_);
    const uint32_t t_sub  = ((uint32_t)lane >> 4) & 1u;
    const uint32_t half   = ((uint32_t)lane >> 3) & 1u;   // 0=vi, 1=ep
    const uint32_t k16    = ((uint32_t)lane & 7u) * 16u;
    const uint32_t g_lane = t_sub * 512u + half * 256u + (uint32_t)u0 * 4u + k16;
    const uint32_t l_lane = t_sub * 256u + half * 128u + k16;
    const uint32_t lds0   = (uint32_t)(uintptr_t)&lds[0][0][0]; // low 32b = LDS offset

    auto issue_chunk = [&](int c, int buf) {
        const uint32_t lbase = lds0 + (uint32_t)buf * (CHUNK * U_ * 4) + l_lane;
        const uint32_t goff  = (uint32_t)c * (CHUNK * 2 * U_ * 4) + g_lane;
#pragma unroll
        for (int j = 0; j < CHUNK / 2; ++j) {
            asm volatile("global_load_async_to_lds_b128 %0, %1, %2"
                         :
                         : "v"(lbase + (uint32_t)j * 512u),
                           "v"(goff  + (uint32_t)j * 1024u),
                           "s"(gbase)
                         : "memory");
        }
    };

    issue_chunk(0, 0);

    float sm = 0.0f, su = 0.0f, sg = 0.0f;
    float* __restrict__ outp = out + (size_t)b * T_ * U_ + u;

    for (int c = 0; c < NCHUNK; ++c) {
        const int buf = c & 1;
        if (c + 1 < NCHUNK) {
            issue_chunk(c + 1, buf ^ 1);
            // 16 newer copies in flight; wait until only those remain ->
            // previous chunk (this buf) is fully resident (in-order completion).
            asm volatile("s_wait_asynccnt 16" ::: "memory");
        } else {
            asm volatile("s_wait_asynccnt 0" ::: "memory");
        }

#pragma unroll 4
        for (int t = 0; t < CHUNK; ++t) {
            const float vi = lds[buf][t][lane];
            const float ep = lds[buf][t][32 + lane];

            // _soil
            const float sdr  = fmaxf(sm + vi - fc_, 0.0f);
            const float inet = vi - sdr;
            // where(sm<tm, ep*sm/tm, ep)*kv == ep*kv*min(sm/tm, 1)
            const float ea   = ep * kv_ * fminf(sm * inv_tm, 1.0f);
            const float ca   = fmaxf(cf_ - cfofc * sm, 0.0f);
            // pow(sm/fc, bt) = exp2(bt * log2(sm/fc)); x=0 -> 0 (matches jnp.power)
            const float pw   = __builtin_amdgcn_exp2f(
                                   bt_ * __builtin_amdgcn_logf(sm * inv_fc));
            const float sp   = inet * pw;

            // _quick + groundwater
            const float qh = ka_ * fmaxf(su - lu_, 0.0f) + kaab * su;
            const float qp = kp_ * su;
            const float qg = kaabag * sg;

            const float d1 = inet + ca - ea - sp;
            const float d2 = sp - ca - qh - qp;
            const float d3 = qp - qg;

            sm = fmaxf(sm + fminf(fmaxf(d1, -CLIPV), CLIPV), 1.0f);
            su = fmaxf(su + fminf(fmaxf(d2, -CLIPV), CLIPV), 0.0f);
            sg = fmaxf(sg + fminf(fmaxf(d3, -CLIPV), CLIPV), 0.0f);

            __builtin_nontemporal_store(sg, outp + (size_t)(c * CHUNK + t) * U_);
        }
    }
}

extern "C" void kernel_launch(void* const* d_in, const int* in_sizes, int n_in,
                              void* d_out, int out_size, void* d_ws, size_t ws_size,
                              hipStream_t stream) {
    const float* inputs = (const float*)d_in[0];
    const float* kv = (const float*)d_in[1];
    const float* fc = (const float*)d_in[2];
    const float* lp = (const float*)d_in[3];
    const float* cf = (const float*)d_in[4];
    const float* bt = (const float*)d_in[5];
    const float* ka = (const float*)d_in[6];
    const float* ab = (const float*)d_in[7];
    const float* lu = (const float*)d_in[8];
    const float* ag = (const float*)d_in[9];
    const float* kp = (const float*)d_in[10];
    float* out = (float*)d_out;

    // 256 single-wave workgroups: one wave per (b, 32-u half); max WGP spread.
    hipLaunchKernelGGL(hbv_scan_kernel, dim3(B_ * 2), dim3(32), 0, stream,
                       inputs, kv, fc, lp, cf, bt, ka, ab, lu, ag, kp, out);
}